// PointCloudProcessor_26980984553514
// MI455X (gfx1250) — compile-verified
//
#include <hip/hip_runtime.h>
#include <stdint.h>

// Problem constants (from the reference)
#define BATCHES 32
#define NPTS    100000
#define DIMS    6
#define KSEL    1024

// FPS kernel geometry
#define TPB     512                 // 16 waves (wave32) per workgroup
#define VEC     4                   // points per thread per vector step
#define VSTEP   (TPB * VEC)         // 2048 points per vector step
#define NVFULL  (NPTS / VSTEP)      // 48 full vector steps (98304 points)
#define NREM    (NPTS - NVFULL * VSTEP)   // 1696 remainder points
#define LDSP    24576               // points cached in LDS (SoA fp32) = 12 vector steps
#define LDSV    (LDSP / VSTEP)      // 12
static_assert(LDSV * VSTEP == LDSP, "LDS cache must cover whole vector steps");

// Normalization kernel geometry
#define NTPB    1024                // one thread per sampled point

typedef float v4f __attribute__((ext_vector_type(4)));

// ---------------------------------------------------------------------------
// CDNA5 async global->LDS copy (ASYNCcnt path, gfx1250)
// ---------------------------------------------------------------------------
__device__ __forceinline__ unsigned lds_off(const void* p) {
  // low 32 bits of a generic pointer into LDS == wave-relative LDS byte offset
  return (unsigned)(uintptr_t)p;
}
__device__ __forceinline__ void async_g2l_b32(unsigned l, const void* g) {
  asm volatile("global_load_async_to_lds_b32 %0, %1, off"
               :: "v"(l), "v"(g) : "memory");
}
__device__ __forceinline__ void async_g2l_b128(unsigned l, const void* g) {
  asm volatile("global_load_async_to_lds_b128 %0, %1, off"
               :: "v"(l), "v"(g) : "memory");
}
__device__ __forceinline__ void wait_async0() {
  asm volatile("s_wait_asynccnt 0" ::: "memory");
}

// ---------------------------------------------------------------------------
// Kernel 0: AoS (B,N,6) -> SoA x[],y[],z[] for coalesced/vectorized hot loop
// ---------------------------------------------------------------------------
__global__ void pack_xyz(const float* __restrict__ pts, float* __restrict__ soa) {
  size_t gid = (size_t)blockIdx.x * blockDim.x + threadIdx.x;
  const size_t BN = (size_t)BATCHES * NPTS;
  if (gid >= BN) return;
  const float* p = pts + gid * DIMS;
  soa[gid]          = p[0];
  soa[BN + gid]     = p[1];
  soa[2 * BN + gid] = p[2];
}

// ---------------------------------------------------------------------------
// Kernel 1: farthest point sampling, one workgroup per batch
// ---------------------------------------------------------------------------
template<bool PACKED>
__global__ __launch_bounds__(TPB, 1)
void fps_kernel(const float* __restrict__ pts,      // AoS (B,N,6)
                const float* __restrict__ soa,      // SoA (3,B*N) or nullptr
                int* idx_out, int idx_stride) {
  extern __shared__ float smem[];                   // 3 * LDSP floats
  float* sx = smem;
  float* sy = smem + LDSP;
  float* sz = smem + 2 * LDSP;
  const v4f* sx4 = (const v4f*)sx;
  const v4f* sy4 = (const v4f*)sy;
  const v4f* sz4 = (const v4f*)sz;
  __shared__ float s_rv[TPB / 32];
  __shared__ int   s_ri[TPB / 32];
  __shared__ int   s_li;

  const int b   = blockIdx.x;
  const int tid = threadIdx.x;
  const size_t BN = (size_t)BATCHES * NPTS;
  const float* gx = PACKED ? (soa + (size_t)b * NPTS) : nullptr;
  const float* gy = PACKED ? (gx + BN) : nullptr;
  const float* gz = PACKED ? (gy + BN) : nullptr;
  const v4f* gx4 = (const v4f*)gx;
  const v4f* gy4 = (const v4f*)gy;
  const v4f* gz4 = (const v4f*)gz;
  const float* ap = pts + (size_t)b * NPTS * DIMS;

  // ---- stage first LDSP points' xyz into LDS via async global->LDS DMA ----
  if (PACKED) {
    #pragma unroll
    for (int pass = 0; pass < (LDSP * 4) / (TPB * 16); ++pass) {
      unsigned off = (unsigned)(pass * TPB * 16 + tid * 16);
      async_g2l_b128(lds_off(sx) + off, (const char*)gx + off);
      async_g2l_b128(lds_off(sy) + off, (const char*)gy + off);
      async_g2l_b128(lds_off(sz) + off, (const char*)gz + off);
    }
  } else {
    for (int i = tid; i < LDSP; i += TPB) {
      const float* p = ap + (size_t)i * DIMS;
      async_g2l_b32(lds_off(sx + i), p + 0);
      async_g2l_b32(lds_off(sy + i), p + 1);
      async_g2l_b32(lds_off(sz + i), p + 2);
    }
  }
  wait_async0();
  __syncthreads();

  // running squared distances, fully register-resident
  v4f dist4[NVFULL];
  #pragma unroll
  for (int j = 0; j < NVFULL; ++j) dist4[j] = (v4f)(1e10f);
  float distt[VEC];
  #pragma unroll
  for (int m = 0; m < VEC; ++m) distt[m] = 1e10f;

  if (tid == 0) idx_out[(size_t)b * idx_stride] = 0;

  float lx = sx[0], ly = sy[0], lz = sz[0];   // first pick = point 0

  for (int k = 1; k < KSEL; ++k) {
    float bestv = -1.0f;
    int   besti = 0;
    v4f lx4 = (v4f)(lx), ly4 = (v4f)(ly), lz4 = (v4f)(lz);

    #pragma unroll
    for (int j = 0; j < NVFULL; ++j) {
      int e = tid + j * TPB;                  // float4 element index
      v4f x, y, z;
      if (j < LDSV)    { x = sx4[e]; y = sy4[e]; z = sz4[e]; }
      else if (PACKED) { x = gx4[e]; y = gy4[e]; z = gz4[e]; }
      else {
        const float* p = ap + (size_t)e * (4 * DIMS);
        x = (v4f){p[0], p[6], p[12], p[18]};
        y = (v4f){p[1], p[7], p[13], p[19]};
        z = (v4f){p[2], p[8], p[14], p[20]};
      }
      // packed f32 math: v_pk_add/mul/fma pairs
      v4f dx = x - lx4, dy = y - ly4, dz = z - lz4;
      v4f d  = dx * dx + dy * dy + dz * dz;
      v4f od = dist4[j];
      float nd0 = fminf(od.x, d.x);
      float nd1 = fminf(od.y, d.y);
      float nd2 = fminf(od.z, d.z);
      float nd3 = fminf(od.w, d.w);
      dist4[j] = (v4f){nd0, nd1, nd2, nd3};
      int ib = e * 4;
      if (nd0 > bestv) { bestv = nd0; besti = ib;     }
      if (nd1 > bestv) { bestv = nd1; besti = ib + 1; }
      if (nd2 > bestv) { bestv = nd2; besti = ib + 2; }
      if (nd3 > bestv) { bestv = nd3; besti = ib + 3; }
    }
    // scalar remainder: points [NVFULL*VSTEP, NPTS)
    #pragma unroll
    for (int m = 0; m < VEC; ++m) {
      int i = NVFULL * VSTEP + tid + m * TPB;
      if (i < NPTS) {
        float x, y, z;
        if (PACKED) { x = gx[i]; y = gy[i]; z = gz[i]; }
        else { const float* p = ap + (size_t)i * DIMS; x = p[0]; y = p[1]; z = p[2]; }
        float dx = x - lx, dy = y - ly, dz = z - lz;
        float d  = fmaf(dz, dz, fmaf(dy, dy, dx * dx));
        float nd = fminf(distt[m], d);
        distt[m] = nd;
        if (nd > bestv) { bestv = nd; besti = i; }
      }
    }

    // wave32 argmax (first-occurrence tie-break to match jnp.argmax)
    #pragma unroll
    for (int off = 16; off > 0; off >>= 1) {
      float ov = __shfl_xor(bestv, off, 32);
      int   oi = __shfl_xor(besti, off, 32);
      if (ov > bestv || (ov == bestv && oi < besti)) { bestv = ov; besti = oi; }
    }
    if ((tid & 31) == 0) { s_rv[tid >> 5] = bestv; s_ri[tid >> 5] = besti; }
    __syncthreads();
    if (tid < (TPB / 32)) {
      bestv = s_rv[tid]; besti = s_ri[tid];
      #pragma unroll
      for (int off = (TPB / 64); off > 0; off >>= 1) {
        float ov = __shfl_xor(bestv, off, 32);
        int   oi = __shfl_xor(besti, off, 32);
        if (ov > bestv || (ov == bestv && oi < besti)) { bestv = ov; besti = oi; }
      }
      if (tid == 0) { s_li = besti; idx_out[(size_t)b * idx_stride + k] = besti; }
    }
    __syncthreads();

    int li = s_li;
    if (li < LDSP)   { lx = sx[li]; ly = sy[li]; lz = sz[li]; }
    else if (PACKED) { lx = gx[li]; ly = gy[li]; lz = gz[li]; }
    else { const float* p = ap + (size_t)li * DIMS; lx = p[0]; ly = p[1]; lz = p[2]; }
  }
}

// ---------------------------------------------------------------------------
// Kernel 2: gather + mean-center + max-abs normalize
// ---------------------------------------------------------------------------
__device__ __forceinline__ float block_sum(float v, float* s, int tid) {
  #pragma unroll
  for (int off = 16; off > 0; off >>= 1) v += __shfl_xor(v, off, 32);
  if ((tid & 31) == 0) s[tid >> 5] = v;
  __syncthreads();
  if (tid < 32) {
    v = s[tid];
    #pragma unroll
    for (int off = 16; off > 0; off >>= 1) v += __shfl_xor(v, off, 32);
    if (tid == 0) s[0] = v;
  }
  __syncthreads();
  v = s[0];
  __syncthreads();
  return v;
}

__device__ __forceinline__ float block_max(float v, float* s, int tid) {
  #pragma unroll
  for (int off = 16; off > 0; off >>= 1) v = fmaxf(v, __shfl_xor(v, off, 32));
  if ((tid & 31) == 0) s[tid >> 5] = v;
  __syncthreads();
  if (tid < 32) {
    v = s[tid];
    #pragma unroll
    for (int off = 16; off > 0; off >>= 1) v = fmaxf(v, __shfl_xor(v, off, 32));
    if (tid == 0) s[0] = v;
  }
  __syncthreads();
  v = s[0];
  __syncthreads();
  return v;
}

__global__ __launch_bounds__(NTPB, 1)
void norm_kernel(const float* __restrict__ pts, const int* idx, int idx_stride,
                 float* out) {
  __shared__ float s[NTPB / 32];
  const int b = blockIdx.x, tid = threadIdx.x;

  int i = idx[(size_t)b * idx_stride + tid];       // may alias out (tier C)
  const float* p = pts + ((size_t)b * NPTS + i) * DIMS;
  float p0 = p[0], p1 = p[1], p2 = p[2], p3 = p[3], p4 = p[4], p5 = p[5];

  const float invK = 1.0f / (float)KSEL;
  float cx = block_sum(p0, s, tid) * invK;
  float cy = block_sum(p1, s, tid) * invK;
  float cz = block_sum(p2, s, tid) * invK;

  float x = p0 - cx, y = p1 - cy, z = p2 - cz;
  float m = fmaxf(fabsf(x), fmaxf(fabsf(y), fabsf(z)));
  float scale = fmaxf(block_max(m, s, tid), 1e-6f);

  float* o = out + ((size_t)b * KSEL + tid) * DIMS;
  o[0] = x / scale; o[1] = y / scale; o[2] = z / scale;
  o[3] = p3; o[4] = p4; o[5] = p5;
}

// ---------------------------------------------------------------------------
extern "C" void kernel_launch(void* const* d_in, const int* in_sizes, int n_in,
                              void* d_out, int out_size, void* d_ws, size_t ws_size,
                              hipStream_t stream) {
  const float* pts = (const float*)d_in[0];
  float* out = (float*)d_out;
  const size_t BN        = (size_t)BATCHES * NPTS;
  const size_t soa_bytes = BN * 3 * sizeof(float);
  const size_t idx_bytes = (size_t)BATCHES * KSEL * sizeof(int);
  const size_t lds_bytes = (size_t)LDSP * 3 * sizeof(float);

  if (d_ws && ws_size >= soa_bytes + idx_bytes) {
    // Tier A: SoA pack in workspace + indices in workspace
    float* soa = (float*)d_ws;
    int*   idx = (int*)((char*)d_ws + soa_bytes);
    pack_xyz<<<(unsigned)((BN + 255) / 256), 256, 0, stream>>>(pts, soa);
    fps_kernel<true><<<BATCHES, TPB, lds_bytes, stream>>>(pts, soa, idx, KSEL);
    norm_kernel<<<BATCHES, NTPB, 0, stream>>>(pts, idx, KSEL, out);
  } else if (d_ws && ws_size >= idx_bytes) {
    // Tier B: AoS reads, indices in workspace
    int* idx = (int*)d_ws;
    fps_kernel<false><<<BATCHES, TPB, lds_bytes, stream>>>(pts, nullptr, idx, KSEL);
    norm_kernel<<<BATCHES, NTPB, 0, stream>>>(pts, idx, KSEL, out);
  } else {
    // Tier C: stage indices inside d_out (per-batch region, overwritten later)
    int* idx = (int*)d_out;
    fps_kernel<false><<<BATCHES, TPB, lds_bytes, stream>>>(pts, nullptr, idx,
                                                           KSEL * DIMS);
    norm_kernel<<<BATCHES, NTPB, 0, stream>>>(pts, idx, KSEL * DIMS, out);
  }
}